// GCN_884763263089
// MI455X (gfx1250) — compile-verified
//
#include <hip/hip_runtime.h>
#include <hip/hip_bf16.h>
#include <math.h>

// ---------------------------------------------------------------------------
// GCN forward on MI455X (gfx1250).
// Dominant cost: x @ W1  (50000x8710 @ 8710x4) -> HBM-bound stream of x
// (1.74 GB -> ~75us floor at 23.3 TB/s). Implemented with
// V_WMMA_F32_16X16X4_F32, one wave32 per 16-row tile, guard-free K loop.
// ---------------------------------------------------------------------------

typedef float v2f __attribute__((ext_vector_type(2)));
typedef float v8f __attribute__((ext_vector_type(8)));

static constexpr int F_IN = 8710;
static constexpr int H1 = 4, H2 = 4, H3 = 2, N_CLS = 70;

// ---------------------------------------------------------------------------
// 1) Degree / normalization:  deg = 1 (self loop) + #in-edges ; dinv = rsqrt
// ---------------------------------------------------------------------------
__global__ void k_fill_deg(float* __restrict__ deg, int n) {
    int i = blockIdx.x * blockDim.x + threadIdx.x;
    if (i < n) deg[i] = 1.0f;
}

__global__ void k_count_deg(const long long* __restrict__ dst,
                            float* __restrict__ deg, int E) {
    int e = blockIdx.x * blockDim.x + threadIdx.x;
    if (e < E) atomicAdd(&deg[(int)dst[e]], 1.0f);
}

__global__ void k_rsqrt_deg(float* __restrict__ deg, int n) {
    int i = blockIdx.x * blockDim.x + threadIdx.x;
    if (i < n) deg[i] = rsqrtf(deg[i]);
}

// ---------------------------------------------------------------------------
// 2) hw = x @ W1 via V_WMMA_F32_16X16X4_F32.
//    A (16x4 f32): lane l holds A[l&15][2*(l>>4) + {0,1}]    (2 VGPRs)
//    B (4x16 f32): lane l holds B[2*(l>>4) + {0,1}][l&15]    (2 VGPRs)
//    C/D (16x16 f32): VGPR r, lane l -> M = r + 8*(l>>4), N = l&15
//    No branches in the loop: all lane masking is done with 0/1 multiplies
//    on clamped (always-valid) addresses.
// ---------------------------------------------------------------------------
__device__ __forceinline__
v8f chunk_wmma(const float* __restrict__ xrow, const float* __restrict__ W1,
               int k, int col, float bmask, v8f acc) {
    v2f a, b;
    a.x = xrow[k];
    a.y = xrow[k + 1];
    b.x = W1[(size_t)k * H1 + col] * bmask;
    b.y = W1[(size_t)(k + 1) * H1 + col] * bmask;
    return __builtin_amdgcn_wmma_f32_16x16x4_f32(
        false, a, false, b, (short)0, acc, false, false);
}

__global__ __launch_bounds__(32)
void k_gemm_xw1_wmma(const float* __restrict__ x,
                     const float* __restrict__ W1,
                     float* __restrict__ hw) {
    const int tile = blockIdx.x;         // 50000/16 = 3125 tiles, exact
    const int lane = threadIdx.x & 31;   // wave32, EXEC all-ones throughout
    const int m    = lane & 15;          // A row within tile / B column
    const int half = lane >> 4;
    const int koff = half * 2;           // this lane's K sub-offset in a chunk
    const int col  = m & 3;              // clamped W1 column (always valid)
    const float bmask = (m < H1) ? 1.0f : 0.0f;   // zero the 12 pad columns

    const float* __restrict__ xrow = x + (size_t)(tile * 16 + m) * F_IN;

    constexpr int K8 = (F_IN / 8) * 8;   // 8704: step-8 main loop bound

    v8f acc0 = {};
    v8f acc1 = {};
    for (int kc = 0; kc < K8; kc += 8) {
        __builtin_prefetch(xrow + kc + 512, 0, 1);   // speculative, no guard
        acc0 = chunk_wmma(xrow, W1, kc + koff,     col, bmask, acc0);
        acc1 = chunk_wmma(xrow, W1, kc + 4 + koff, col, bmask, acc1);
    }
    // leftover full chunk: k = 8704..8707
    acc0 = chunk_wmma(xrow, W1, K8 + koff, col, bmask, acc0);

    // 2-wide tail (k = 8708, 8709): carried by half==0 lanes; half==1 lanes
    // are zeroed via mask and read from a clamped, always-in-bounds address.
    {
        const float hm  = (half == 0) ? 1.0f : 0.0f;
        const float bm2 = bmask * hm;
        const int   kt  = half ? (F_IN - 6) : (F_IN - 2);  // 8704 / 8708
        v2f a, b;
        a.x = xrow[kt] * hm;
        a.y = xrow[kt + 1] * hm;
        b.x = W1[(size_t)kt * H1 + col] * bm2;
        b.y = W1[(size_t)(kt + 1) * H1 + col] * bm2;
        acc1 = __builtin_amdgcn_wmma_f32_16x16x4_f32(
            false, a, false, b, (short)0, acc1, false, false);
    }

    const v8f acc = acc0 + acc1;

    // store columns 0..3 of the 16x16 D tile
    if (m < H1) {
#pragma unroll
        for (int r = 0; r < 8; ++r) {
            const int row = tile * 16 + r + 8 * half;
            hw[(size_t)row * H1 + m] = acc[r];
        }
    }
}

// ---------------------------------------------------------------------------
// 3) Message passing:  agg = b + dinv[i]^2 * hw[i]  (self loop), then
//    agg[dst] += dinv[src]*dinv[dst] * hw[src]  over all edges.
// ---------------------------------------------------------------------------
template <int H>
__global__ void k_agg_init(const float* __restrict__ hw,
                           const float* __restrict__ dinv,
                           const float* __restrict__ bias,
                           float* __restrict__ agg, int n) {
    int i = blockIdx.x * blockDim.x + threadIdx.x;
    if (i >= n) return;
    const float w = dinv[i] * dinv[i];
#pragma unroll
    for (int f = 0; f < H; ++f)
        agg[(size_t)i * H + f] = bias[f] + hw[(size_t)i * H + f] * w;
}

template <int H>
__global__ void k_edge_msg(const long long* __restrict__ src,
                           const long long* __restrict__ dst,
                           const float* __restrict__ hw,
                           const float* __restrict__ dinv,
                           float* __restrict__ agg, int E) {
    int e = blockIdx.x * blockDim.x + threadIdx.x;
    if (e >= E) return;
    const int s = (int)src[e];
    const int d = (int)dst[e];
    const float w = dinv[s] * dinv[d];
#pragma unroll
    for (int f = 0; f < H; ++f)
        atomicAdd(&agg[(size_t)d * H + f], hw[(size_t)s * H + f] * w);
}

// ---------------------------------------------------------------------------
// 4) Fused ReLU + next-layer linear (tiny: 4x4 / 4x2), per node.
// ---------------------------------------------------------------------------
template <int HIN, int HOUT>
__global__ void k_relu_transform(const float* __restrict__ agg,
                                 const float* __restrict__ W,
                                 float* __restrict__ hwout, int n) {
    int i = blockIdx.x * blockDim.x + threadIdx.x;
    if (i >= n) return;
    float h[HIN];
#pragma unroll
    for (int f = 0; f < HIN; ++f) h[f] = fmaxf(agg[(size_t)i * HIN + f], 0.0f);
#pragma unroll
    for (int g = 0; g < HOUT; ++g) {
        float s = 0.0f;
#pragma unroll
        for (int f = 0; f < HIN; ++f) s = fmaf(h[f], W[f * HOUT + g], s);
        hwout[(size_t)i * HOUT + g] = s;
    }
}

// ---------------------------------------------------------------------------
// 5) Final: h3 = relu(agg) (2 wide) -> d_out tail + scratch; then classifier
// ---------------------------------------------------------------------------
__global__ void k_final_h(const float* __restrict__ agg,
                          float* __restrict__ h3scratch,
                          float* __restrict__ out_h_tail, int n) {
    int i = blockIdx.x * blockDim.x + threadIdx.x;
    if (i >= n) return;
    const float a = fmaxf(agg[(size_t)i * 2 + 0], 0.0f);
    const float b = fmaxf(agg[(size_t)i * 2 + 1], 0.0f);
    h3scratch[(size_t)i * 2 + 0] = a;
    h3scratch[(size_t)i * 2 + 1] = b;
    out_h_tail[(size_t)i * 2 + 0] = a;
    out_h_tail[(size_t)i * 2 + 1] = b;
}

__global__ void k_classifier(const float* __restrict__ h3,
                             const float* __restrict__ Wc,
                             const float* __restrict__ bc,
                             float* __restrict__ out, int n) {
    int t = blockIdx.x * blockDim.x + threadIdx.x;
    if (t >= n * N_CLS) return;
    const int i = t / N_CLS;
    const int c = t - i * N_CLS;
    out[t] = fmaf(h3[(size_t)i * 2 + 0], Wc[c],
             fmaf(h3[(size_t)i * 2 + 1], Wc[N_CLS + c], bc[c]));
}

// ---------------------------------------------------------------------------
// Driver
// ---------------------------------------------------------------------------
extern "C" void kernel_launch(void* const* d_in, const int* in_sizes, int n_in,
                              void* d_out, int out_size, void* d_ws, size_t ws_size,
                              hipStream_t stream) {
    const float*     x    = (const float*)d_in[0];
    const long long* eidx = (const long long*)d_in[1];   // int64, shape (2, E)
    const float*     W1   = (const float*)d_in[2];
    const float*     b1   = (const float*)d_in[3];
    const float*     W2   = (const float*)d_in[4];
    const float*     b2   = (const float*)d_in[5];
    const float*     W3   = (const float*)d_in[6];
    const float*     b3   = (const float*)d_in[7];
    const float*     Wc   = (const float*)d_in[8];
    const float*     bc   = (const float*)d_in[9];

    const int N = in_sizes[0] / F_IN;        // 50000
    const int E = in_sizes[1] / 2;           // 1600000
    const long long* src = eidx;
    const long long* dst = eidx + E;

    float* out_logits = (float*)d_out;                      // [N, 70]
    float* out_h      = out_logits + (size_t)N * N_CLS;     // [N, 2] tail

    // workspace layout
    float* dinv = (float*)d_ws;            // N
    float* hw   = dinv + N;                // N*4 (max width)
    float* agg  = hw + (size_t)N * 4;      // N*4

    const int T   = 256;
    const int gN  = (N + T - 1) / T;
    const int gE  = (E + T - 1) / T;
    const int gNC = (N * N_CLS + T - 1) / T;

    // --- normalization -----------------------------------------------------
    k_fill_deg <<<gN, T, 0, stream>>>(dinv, N);
    k_count_deg<<<gE, T, 0, stream>>>(dst, dinv, E);
    k_rsqrt_deg<<<gN, T, 0, stream>>>(dinv, N);

    // --- layer 1: WMMA GEMM + propagate ------------------------------------
    k_gemm_xw1_wmma<<<N / 16, 32, 0, stream>>>(x, W1, hw);
    k_agg_init<H1><<<gN, T, 0, stream>>>(hw, dinv, b1, agg, N);
    k_edge_msg<H1><<<gE, T, 0, stream>>>(src, dst, hw, dinv, agg, E);

    // --- layer 2: relu + 4x4 transform + propagate -------------------------
    k_relu_transform<H1, H2><<<gN, T, 0, stream>>>(agg, W2, hw, N);
    k_agg_init<H2><<<gN, T, 0, stream>>>(hw, dinv, b2, agg, N);
    k_edge_msg<H2><<<gE, T, 0, stream>>>(src, dst, hw, dinv, agg, E);

    // --- layer 3: relu + 4x2 transform + propagate -------------------------
    k_relu_transform<H2, H3><<<gN, T, 0, stream>>>(agg, W3, hw, N);
    k_agg_init<H3><<<gN, T, 0, stream>>>(hw, dinv, b3, agg, N);
    k_edge_msg<H3><<<gE, T, 0, stream>>>(src, dst, hw, dinv, agg, E);

    // --- outputs: h3 (relu) + classifier -----------------------------------
    k_final_h   <<<gN,  T, 0, stream>>>(agg, hw, out_h, N);
    k_classifier<<<gNC, T, 0, stream>>>(hw, Wc, bc, out_logits, N);
}